// SAE_38190849196168
// MI455X (gfx1250) — compile-verified
//
#include <hip/hip_runtime.h>
#include <hip/hip_bf16.h>

// ---------------------------------------------------------------------------
// SAE forward:  out = TopK32(relu((x - b_dec) @ W_enc^T + b_enc)) @ W_dec + b_dec
// B=4096, D=256, H=8192, K=32.
//
// Pass 0: beff[h] = b_enc[h] - dot(W_enc[h,:], b_dec)   (folds b_dec into bias
//         so the GEMM operands are pure copies -> async DMA-able)
// Pass 1: fp32 WMMA GEMM pre = relu(x @ W_enc^T + beff), K streamed through
//         double-buffered LDS filled with GLOBAL_LOAD_ASYNC_TO_LDS_B128
//         (ASYNCcnt) so the copy of chunk k+1 overlaps the 64 WMMAs of chunk k.
// Pass 2: per-row exact top-32 + sparse decode.
// ---------------------------------------------------------------------------

typedef float v2f __attribute__((ext_vector_type(2)));
typedef float v8f __attribute__((ext_vector_type(8)));

#define LSTRIDE 68   // 64 + 4 pad floats; 68 % 64 == 4 -> conflict-free frag reads

// Async global->LDS 128-bit copy. The LDS operand is the 32-bit LDS byte
// offset: generic pointers to __shared__ carry the aperture in the high dword
// and the LDS offset in the low dword, so truncation yields the VDST value.
__device__ __forceinline__ void async_copy_b128(const float* g, const float* l)
{
    unsigned loff = (unsigned)(unsigned long long)(const void*)l;
    asm volatile("global_load_async_to_lds_b128 %0, %1, off"
                 :: "v"(loff), "v"(g) : "memory");
}
__device__ __forceinline__ void wait_async0()
{
    asm volatile("s_wait_asynccnt 0x0" ::: "memory");
}

// ---------------------------------------------------------------------------
// Pass 0: effective encoder bias.
// ---------------------------------------------------------------------------
__global__ __launch_bounds__(256) void sae_beff(
    const float* __restrict__ W_enc, const float* __restrict__ b_enc,
    const float* __restrict__ b_dec, float* __restrict__ beff, int H, int D)
{
    int h = blockIdx.x * 256 + threadIdx.x;
    if (h >= H) return;
    float acc = b_enc[h];
    const float* wrow = W_enc + (long)h * D;
    for (int k = 0; k < D; k += 4) {
        float4 w = *(const float4*)(wrow + k);
        float4 b = *(const float4*)(b_dec + k);
        acc -= w.x * b.x + w.y * b.y + w.z * b.z + w.w * b.w;
    }
    beff[h] = acc;
}

// ---------------------------------------------------------------------------
// Pass 1: pre[m, n] = relu( sum_k x[m,k] * W_enc[n,k] + beff[n] )
// Workgroup tile: 64(M) x 128(N); K in chunks of 64 via double-buffered LDS.
// 8 waves as 2(M) x 4(N); each wave owns a 32x32 subtile (2x2 WMMA tiles,
// K-step 4 via V_WMMA_F32_16X16X4_F32).
// ---------------------------------------------------------------------------
__global__ __launch_bounds__(256) void sae_encode_wmma(
    const float* __restrict__ x, const float* __restrict__ W_enc,
    const float* __restrict__ beff, float* __restrict__ pre,
    int row0, int H, int D)
{
    __shared__ float As[2][64 * LSTRIDE];    //  34.8 KB
    __shared__ float Bs[2][128 * LSTRIDE];   //  69.6 KB  (104.4 KB total)

    const int tid  = threadIdx.x;
    const int lane = tid & 31;
    const int wave = tid >> 5;
    const int wm   = wave >> 2;          // 0..1 : wave M position
    const int wn   = wave & 3;           // 0..3 : wave N position
    const int ll   = lane & 15;          // fragment row (M for A, N for B)
    const int lh   = lane >> 4;          // half-wave: K offset +2 (A/B), M offset +8 (C/D)

    const int nblk = blockIdx.x;         // H/128 blocks
    const int mblk = blockIdx.y;         // rows/64 blocks
    const long grow0 = (long)row0 + (long)mblk * 64;

    const int NC = D >> 6;               // K chunks of 64

    // thread's fixed staging coordinates (row pair advances by 16 per iter)
    v8f acc[2][2] = {};

    // ---- issue async copies for one 64-wide K chunk into buffer `buf`
    auto stage = [&](int kc, int buf) {
        float* Ad = &As[buf][0];
        #pragma unroll
        for (int i = 0; i < 4; ++i) {
            int lin = i * 256 + tid;
            int r = lin >> 4, c = (lin & 15) * 4;
            async_copy_b128(x + (grow0 + r) * (long)D + kc + c,
                            Ad + r * LSTRIDE + c);
        }
        float* Bd = &Bs[buf][0];
        #pragma unroll
        for (int i = 0; i < 8; ++i) {
            int lin = i * 256 + tid;
            int r = lin >> 4, c = (lin & 15) * 4;
            async_copy_b128(W_enc + ((long)nblk * 128 + r) * (long)D + kc + c,
                            Bd + r * LSTRIDE + c);
        }
    };

    stage(0, 0);
    for (int ci = 0; ci < NC; ++ci) {
        const int cur = ci & 1;
        wait_async0();                  // this wave's copies for chunk ci done
        __syncthreads();                // everyone's chunk-ci data visible
        if (ci + 1 < NC)
            stage((ci + 1) << 6, cur ^ 1);   // overlaps with compute below

        const float* Ab = &As[cur][0];
        const float* Bb = &Bs[cur][0];
        #pragma unroll
        for (int kk = 0; kk < 16; ++kk) {
            const int ko = kk * 4 + lh * 2;   // lanes 16-31 carry K+2 (ISA 16x4 layout)
            v2f a0 = *(const v2f*)(&Ab[(wm * 32 +      ll) * LSTRIDE + ko]);
            v2f a1 = *(const v2f*)(&Ab[(wm * 32 + 16 + ll) * LSTRIDE + ko]);
            v2f b0 = *(const v2f*)(&Bb[(wn * 32 +      ll) * LSTRIDE + ko]);
            v2f b1 = *(const v2f*)(&Bb[(wn * 32 + 16 + ll) * LSTRIDE + ko]);
            acc[0][0] = __builtin_amdgcn_wmma_f32_16x16x4_f32(false, a0, false, b0,
                            (short)0, acc[0][0], false, false);
            acc[0][1] = __builtin_amdgcn_wmma_f32_16x16x4_f32(false, a0, false, b1,
                            (short)0, acc[0][1], false, false);
            acc[1][0] = __builtin_amdgcn_wmma_f32_16x16x4_f32(false, a1, false, b0,
                            (short)0, acc[1][0], false, false);
            acc[1][1] = __builtin_amdgcn_wmma_f32_16x16x4_f32(false, a1, false, b1,
                            (short)0, acc[1][1], false, false);
        }
        __syncthreads();                // all reads of buf `cur` finished
    }

    // ---- epilogue: + beff, relu, store to workspace
    // C/D layout: VGPR j -> row j (lanes 0-15) / row j+8 (lanes 16-31), col = ll
    #pragma unroll
    for (int nt = 0; nt < 2; ++nt) {
        int colg = nblk * 128 + wn * 32 + nt * 16 + ll;
        float be = beff[colg];
        #pragma unroll
        for (int mt = 0; mt < 2; ++mt) {
            #pragma unroll
            for (int j = 0; j < 8; ++j) {
                int mloc = mblk * 64 + wm * 32 + mt * 16 + lh * 8 + j;
                float v = acc[mt][nt][j] + be;
                v = v > 0.f ? v : 0.f;
                pre[(long)mloc * H + colg] = v;
            }
        }
    }
}

// ---------------------------------------------------------------------------
// Pass 2: exact top-32 per row (lowest-index tie-break == lax.top_k) + decode.
// One 256-thread workgroup per batch row. Dynamic LDS holds the H-float row.
// ---------------------------------------------------------------------------
__global__ __launch_bounds__(256) void sae_topk_decode(
    const float* __restrict__ pre, const float* __restrict__ W_dec,
    const float* __restrict__ b_dec, float* __restrict__ out,
    int row0, int H, int D)
{
    extern __shared__ float srow[];      // H floats
    __shared__ float rv[256];
    __shared__ int   ri[256];
    __shared__ float topv[32];
    __shared__ int   topi[32];

    const int tid  = threadIdx.x;
    const int lrow = blockIdx.x;
    const float* prow = pre + (long)lrow * H;

    for (int i = tid; i < H; i += 256) srow[i] = prow[i];
    __syncthreads();

    for (int k = 0; k < 32; ++k) {
        float bv = -1.0f;   // all candidates are >= 0 after relu
        int   bi = 0x7fffffff;
        for (int i = tid; i < H; i += 256) {
            float v = srow[i];
            if (v > bv) { bv = v; bi = i; }   // strict '>' keeps lowest index per thread
        }
        rv[tid] = bv; ri[tid] = bi;
        __syncthreads();
        for (int s = 128; s > 0; s >>= 1) {
            if (tid < s) {
                float ov = rv[tid + s]; int oi = ri[tid + s];
                if (ov > rv[tid] || (ov == rv[tid] && oi < ri[tid])) {
                    rv[tid] = ov; ri[tid] = oi;
                }
            }
            __syncthreads();
        }
        if (tid == 0) {
            topv[k] = rv[0]; topi[k] = ri[0];
            srow[ri[0]] = -1.0f;             // remove from further selection
        }
        __syncthreads();
    }

    // decode: out[row, d] = b_dec[d] + sum_k topv[k] * W_dec[topi[k], d]
    for (int dd = tid; dd < D; dd += 256) {
        float acc = b_dec[dd];
        #pragma unroll 8
        for (int k = 0; k < 32; ++k)
            acc += topv[k] * W_dec[(long)topi[k] * D + dd];
        out[((long)row0 + lrow) * D + dd] = acc;
    }
}

// ---------------------------------------------------------------------------
extern "C" void kernel_launch(void* const* d_in, const int* in_sizes, int n_in,
                              void* d_out, int out_size, void* d_ws, size_t ws_size,
                              hipStream_t stream) {
    const float* x     = (const float*)d_in[0];
    const float* W_enc = (const float*)d_in[1];
    const float* b_enc = (const float*)d_in[2];
    const float* W_dec = (const float*)d_in[3];
    const float* b_dec = (const float*)d_in[4];
    float* out = (float*)d_out;

    const int D = in_sizes[4];            // 256
    const int H = in_sizes[2];            // 8192
    const int B = in_sizes[0] / D;        // 4096

    // workspace layout: [ beff : H floats ][ pre-activation slab ]
    float* beff = (float*)d_ws;
    float* preb = beff + H;

    const size_t row_bytes = (size_t)H * sizeof(float);
    size_t avail = (ws_size > row_bytes) ? (ws_size - row_bytes) : 0;
    long max_rows = (long)(avail / row_bytes);
    max_rows = (max_rows / 64) * 64;
    if (max_rows < 64) max_rows = 64;     // assume workspace covers >= one slab
    int slab = (int)((max_rows < B) ? max_rows : B);

    sae_beff<<<(H + 255) / 256, 256, 0, stream>>>(W_enc, b_enc, b_dec, beff, H, D);

    for (int row0 = 0; row0 < B; row0 += slab) {
        int rows = B - row0; if (rows > slab) rows = slab;
        dim3 g1(H / 128, rows / 64);
        sae_encode_wmma<<<g1, 256, 0, stream>>>(x, W_enc, beff, preb, row0, H, D);
        sae_topk_decode<<<rows, 256, row_bytes, stream>>>(preb, W_dec, b_dec, out,
                                                          row0, H, D);
    }
}